// ToRGB_ATT2_40261023433216
// MI455X (gfx1250) — compile-verified
//
#include <hip/hip_runtime.h>

typedef __attribute__((ext_vector_type(2))) float v2f;
typedef __attribute__((ext_vector_type(8))) float v8f;

#define NB   8
#define CIN  512
#define HH   128
#define WW   128
#define HW   (HH * WW)

// workspace layout in floats
#define WS_WMOD   0                           // [8][3][512]    = 12288
#define WS_SKIPUP (WS_WMOD + NB * 3 * CIN)    // [8][3][16384]  = 393216
#define WS_ENERGY (WS_SKIPUP + NB * 3 * HW)   // [8][16][1024]  = 131072
#define WS_SUMSQ  (WS_ENERGY + NB * 16 * 1024)// [8]

#define INV_SQRT512 0.04419417382415922f

// ---------------------------------------------------------------------------
// Phase A: modulated 1x1 weights  wmod[b][o][c] = conv_w[o][c]/sqrt(512) * s[b][c]
//          s[b][c] = dot(style[b], mod_w[c]) / sqrt(512) + mod_b[c]
// ---------------------------------------------------------------------------
__global__ void k_modulate(const float* __restrict__ style,
                           const float* __restrict__ conv_w,
                           const float* __restrict__ mod_w,
                           const float* __restrict__ mod_b,
                           float* __restrict__ wmod) {
  int b = blockIdx.x;
  int c = threadIdx.x;                 // 512 threads
  const float* st = style + b * CIN;
  const float* mw = mod_w + c * CIN;
  float acc = 0.0f;
  for (int j = 0; j < CIN; ++j) acc += st[j] * mw[j];
  float s  = acc * INV_SQRT512 + mod_b[c];
  float cs = INV_SQRT512 * s;          // fold conv_scale
  for (int o = 0; o < 3; ++o)
    wmod[(b * 3 + o) * CIN + c] = conv_w[o * CIN + c] * cs;
}

// ---------------------------------------------------------------------------
// Phase B: upfirdn2d(skip, k, up=2, pad=(2,1)) -> skip_up [8,3,128,128]
// 2x2 effective taps per output pixel, kernel = outer([1,3,3,1])/16
// ---------------------------------------------------------------------------
__global__ void k_upsample(const float* __restrict__ skip,
                           float* __restrict__ skip_up) {
  int idx = blockIdx.x * blockDim.x + threadIdx.x;
  if (idx >= NB * 3 * HW) return;
  int x  = idx & 127;
  int y  = (idx >> 7) & 127;
  int bc = idx >> 14;
  const float* sp = skip + bc * 64 * 64;
  const float v[4] = {1.0f, 3.0f, 3.0f, 1.0f};

  int iy[2], sy[2], ix[2], sx[2];
  if (y & 1) { iy[0] = 1; sy[0] = (y - 1) >> 1; iy[1] = 3; sy[1] = (y + 1) >> 1; }
  else       { iy[0] = 0; sy[0] = (y - 2) / 2;  iy[1] = 2; sy[1] = y >> 1; }
  if (x & 1) { ix[0] = 1; sx[0] = (x - 1) >> 1; ix[1] = 3; sx[1] = (x + 1) >> 1; }
  else       { ix[0] = 0; sx[0] = (x - 2) / 2;  ix[1] = 2; sx[1] = x >> 1; }

  float acc = 0.0f;
  for (int a = 0; a < 2; ++a)
    for (int c = 0; c < 2; ++c)
      if (sy[a] >= 0 && sy[a] < 64 && sx[c] >= 0 && sx[c] < 64)
        acc += v[iy[a]] * v[ix[c]] * sp[sy[a] * 64 + sx[c]];
  skip_up[idx] = acc * (1.0f / 16.0f);
}

// ---------------------------------------------------------------------------
// Phase C: 3x3 conv on attention maps, per-4x4-block L2 norms, energy,
//          per-batch sum of energy^2 (for the global L2 normalization).
// One thread per 4x4 block (8 * 1024 blocks).
// ---------------------------------------------------------------------------
__global__ void k_energy(const float* __restrict__ att0,
                         const float* __restrict__ att1,
                         const float* __restrict__ q_w,
                         const float* __restrict__ q_b,
                         float* __restrict__ energy,
                         float* __restrict__ sumsq) {
  int t = blockIdx.x * blockDim.x + threadIdx.x;
  if (t >= NB * 1024) return;
  int b = t >> 10;
  int L = t & 1023;
  int by = L >> 5, bx = L & 31;
  float qb = q_b[0];

  float pq[16], pk[16];
  for (int e = 0; e < 16; ++e) {
    int y = by * 4 + (e >> 2);
    int x = bx * 4 + (e & 3);
    float aq = qb, ak = qb;
    for (int ch = 0; ch < 3; ++ch) {
      const float* p0 = att0 + (b * 3 + ch) * HW;
      const float* p1 = att1 + (b * 3 + ch) * HW;
      for (int ky = 0; ky < 3; ++ky) {
        int yy = y + ky - 1;
        if (yy < 0 || yy >= HH) continue;
        for (int kx = 0; kx < 3; ++kx) {
          int xx = x + kx - 1;
          if (xx < 0 || xx >= WW) continue;
          float wv = q_w[(ch * 3 + ky) * 3 + kx];
          aq += wv * p0[yy * WW + xx];
          ak += wv * p1[yy * WW + xx];
        }
      }
    }
    pq[e] = aq; pk[e] = ak;
  }
  float nq = 0.0f, nk = 0.0f;
  for (int e = 0; e < 16; ++e) { nq += pq[e] * pq[e]; nk += pk[e] * pk[e]; }
  nq = fmaxf(sqrtf(nq), 1e-12f);
  nk = fmaxf(sqrtf(nk), 1e-12f);
  float ss = 0.0f;
  for (int e = 0; e < 16; ++e) {
    float ev = (pq[e] / nq) * (pk[e] / nk);
    energy[b * 16384 + e * 1024 + L] = ev;
    ss += ev * ev;
  }
  atomicAdd(&sumsq[b], ss);
}

// ---------------------------------------------------------------------------
// Phase D: main WMMA contraction + fused attention blend.
// Per wave: 4 tiles of 16 pixels (64 contiguous pixels), K=512 in steps of 4
// via V_WMMA_F32_16X16X4_F32.  M rows 0..2 = output channels (row 3..15 zero).
// ---------------------------------------------------------------------------
__device__ __forceinline__ void epilogue(v8f acc, int b, int p, float att,
                                         float bi0, float bi1, float bi2,
                                         const float* __restrict__ skip_up,
                                         float* __restrict__ out) {
  float om = 1.0f - att;
  int base = b * 3 * HW + p;
  out[base]          = (acc[0] + bi0) * att + skip_up[base] * om;
  out[base + HW]     = (acc[1] + bi1) * att + skip_up[base + HW] * om;
  out[base + 2 * HW] = (acc[2] + bi2) * att + skip_up[base + 2 * HW] * om;
}

__global__ __launch_bounds__(256) void k_main(const float* __restrict__ input,
                                              const float* __restrict__ bias,
                                              const float* __restrict__ wmod,
                                              const float* __restrict__ skip_up,
                                              const float* __restrict__ energy,
                                              const float* __restrict__ sumsq,
                                              float* __restrict__ out) {
  __shared__ float lw[4 * CIN];        // rows 0..2 = wmod, row 3 = zeros
  int b   = blockIdx.x >> 5;           // 32 blocks per batch
  int blk = blockIdx.x & 31;

  for (int i = threadIdx.x; i < 4 * CIN; i += blockDim.x)
    lw[i] = (i < 3 * CIN) ? wmod[b * 3 * CIN + i] : 0.0f;
  __syncthreads();

  int wave = threadIdx.x >> 5;
  int lane = threadIdx.x & 31;
  int half = lane >> 4;
  int m    = lane & 15;                // A row / B-D column within half
  int mrow = (m < 3) ? m : 3;          // rows >=3 read the zero row
  int pix_base = (blk * 8 + wave) * 64;

  const float* inb = input + (size_t)b * CIN * HW;

  v8f acc0 = {}, acc1 = {}, acc2 = {}, acc3 = {};
#pragma unroll 2
  for (int k = 0; k < CIN; k += 4) {
    int c0 = k + 2 * half;             // this half-wave covers K = c0, c0+1
    v2f a;
    a.x = lw[mrow * CIN + c0];
    a.y = lw[mrow * CIN + c0 + 1];
    const float* pb = inb + (size_t)c0 * HW + pix_base + m;
    v2f b0, b1, b2, b3;
    b0.x = pb[0];  b0.y = pb[HW];
    b1.x = pb[16]; b1.y = pb[HW + 16];
    b2.x = pb[32]; b2.y = pb[HW + 32];
    b3.x = pb[48]; b3.y = pb[HW + 48];
    acc0 = __builtin_amdgcn_wmma_f32_16x16x4_f32(false, a, false, b0, (short)0, acc0, false, false);
    acc1 = __builtin_amdgcn_wmma_f32_16x16x4_f32(false, a, false, b1, (short)0, acc1, false, false);
    acc2 = __builtin_amdgcn_wmma_f32_16x16x4_f32(false, a, false, b2, (short)0, acc2, false, false);
    acc3 = __builtin_amdgcn_wmma_f32_16x16x4_f32(false, a, false, b3, (short)0, acc3, false, false);
  }

  if (half == 0) {                     // lanes 0..15 hold M=0..7 in vgprs 0..7
    float scale = 4.0f / fmaxf(sqrtf(sumsq[b]), 1e-12f);
    float bi0 = bias[0], bi1 = bias[1], bi2 = bias[2];
#pragma unroll
    for (int t = 0; t < 4; ++t) {
      int p = pix_base + t * 16 + m;
      int y = p >> 7, x = p & 127;
      int L = (y >> 2) * 32 + (x >> 2);
      int e = (y & 3) * 4 + (x & 3);
      float att = energy[b * 16384 + e * 1024 + L] * scale;
      v8f acc = (t == 0) ? acc0 : (t == 1) ? acc1 : (t == 2) ? acc2 : acc3;
      epilogue(acc, b, p, att, bi0, bi1, bi2, skip_up, out);
    }
  }
}

// ---------------------------------------------------------------------------
extern "C" void kernel_launch(void* const* d_in, const int* in_sizes, int n_in,
                              void* d_out, int out_size, void* d_ws, size_t ws_size,
                              hipStream_t stream) {
  const float* input  = (const float*)d_in[0];
  const float* style  = (const float*)d_in[1];
  const float* skip   = (const float*)d_in[2];
  const float* att0   = (const float*)d_in[3];
  const float* att1   = (const float*)d_in[4];
  const float* conv_w = (const float*)d_in[5];
  const float* mod_w  = (const float*)d_in[6];
  const float* mod_b  = (const float*)d_in[7];
  const float* bias   = (const float*)d_in[8];
  const float* q_w    = (const float*)d_in[9];
  const float* q_b    = (const float*)d_in[10];
  (void)in_sizes; (void)n_in; (void)out_size; (void)ws_size;

  float* ws      = (float*)d_ws;
  float* wmod    = ws + WS_WMOD;
  float* skip_up = ws + WS_SKIPUP;
  float* energy  = ws + WS_ENERGY;
  float* sumsq   = ws + WS_SUMSQ;
  float* out     = (float*)d_out;

  hipMemsetAsync(sumsq, 0, NB * sizeof(float), stream);

  k_modulate<<<NB, CIN, 0, stream>>>(style, conv_w, mod_w, mod_b, wmod);
  k_upsample<<<(NB * 3 * HW + 255) / 256, 256, 0, stream>>>(skip, skip_up);
  k_energy<<<(NB * 1024 + 127) / 128, 128, 0, stream>>>(att0, att1, q_w, q_b,
                                                        energy, sumsq);
  k_main<<<256, 256, 0, stream>>>(input, bias, wmod, skip_up, energy, sumsq, out);
}